// GNN_70411693851070
// MI455X (gfx1250) — compile-verified
//
#include <hip/hip_runtime.h>
#include <math.h>

// ---------------------------------------------------------------------------
// GNN attention block for MI455X (gfx1250, wave32).
// GEMMs: V_WMMA_F32_16X16X4_F32, register-blocked (multi-subtile per wave)
// and software-pipelined so fragment loads overlap WMMA chains (asm shows
// s_clause-grouped loads + partial s_wait_loadcnt).
// Scores kernel: native v_tanh_f32 when available, else 3-op exp-based tanh;
// LDS operands read as b128.
// ---------------------------------------------------------------------------

#define S 256
#define B 8
#define H 512
#define NITER 4

typedef __attribute__((ext_vector_type(2))) float v2f;
typedef __attribute__((ext_vector_type(8))) float v8f;

#if __has_builtin(__builtin_amdgcn_tanhf)
static __device__ __forceinline__ float fast_tanh(float x) {
  return __builtin_amdgcn_tanhf(x);
}
#else
// tanh(x) = 1 - 2/(exp(2x)+1): v_exp + v_rcp + fma; saturates correctly at +-inf
static __device__ __forceinline__ float fast_tanh(float x) {
  float e = __expf(2.0f * x);
  return 1.0f - 2.0f / (e + 1.0f);
}
#endif

static __device__ __forceinline__ v8f wmma_f32(v2f a, v2f b, v8f c) {
  // 8 args: (neg_a, A, neg_b, B, c_mod, C, reuse_a, reuse_b)
  return __builtin_amdgcn_wmma_f32_16x16x4_f32(false, a, false, b, (short)0, c,
                                               false, false);
}

static __device__ __forceinline__ float warpMax(float v) {
  #pragma unroll
  for (int o = 16; o > 0; o >>= 1) v = fmaxf(v, __shfl_xor(v, o, 32));
  return v;
}
static __device__ __forceinline__ float warpSum(float v) {
  #pragma unroll
  for (int o = 16; o > 0; o >>= 1) v += __shfl_xor(v, o, 32);
  return v;
}

// store one 16x16 C tile from the v8f accumulator (+ optional bias per col)
static __device__ __forceinline__ void store_tile(float* C, int ldc, int m0,
                                                  int n0, int r, int g, v8f acc,
                                                  float bv) {
  float* cp = C + (size_t)m0 * ldc + n0 + r;
  #pragma unroll
  for (int v = 0; v < 8; ++v) cp[(size_t)(v + 8 * g) * ldc] = acc[v] + bv;
}

// ---- layout change (S,B,H) -> (B,S,H), vectorized ------------------------
__global__ __launch_bounds__(256) void transpose_in(const float* __restrict__ in,
                                                    float* __restrict__ hid) {
  size_t idx = (size_t)blockIdx.x * blockDim.x + threadIdx.x; // over B*S*H/4
  int hv = (int)(idx % (H / 4));
  size_t bi = idx / (H / 4);
  int i = (int)(bi % S);
  int b = (int)(bi / S);
  const float4* src = (const float4*)(in + ((size_t)i * B + b) * H) + hv;
  float4* dst = (float4*)(hid + ((size_t)b * S + i) * H) + hv;
  *dst = *src;
}

__global__ __launch_bounds__(256) void transpose_out(const float* __restrict__ hid,
                                                     float* __restrict__ out) {
  size_t idx = (size_t)blockIdx.x * blockDim.x + threadIdx.x;
  int hv = (int)(idx % (H / 4));
  size_t bi = idx / (H / 4);
  int i = (int)(bi % S);
  int b = (int)(bi / S);
  const float4* src = (const float4*)(hid + ((size_t)b * S + i) * H) + hv;
  float4* dst = (float4*)(out + ((size_t)i * B + b) * H) + hv;
  *dst = *src;
}

// ---- C[M,N] = X[M,K] @ W[N,K]^T + bias; 64x16 C-strip per wave -----------
// One B(weight) fragment reused by 4 M-subtiles; 1-deep pipeline.
__global__ __launch_bounds__(256) void gemm_xwT(const float* __restrict__ X,
                                                const float* __restrict__ W,
                                                const float* __restrict__ bias,
                                                float* __restrict__ C,
                                                int M, int N, int K) {
  int wave = threadIdx.x >> 5, lane = threadIdx.x & 31;
  int tilesN = N >> 4;
  int tile = blockIdx.x * 8 + wave;          // over (M/64)*(N/16)
  int tmb = tile / tilesN, tn = tile - tmb * tilesN;
  if (tmb * 64 >= M) return;                 // wave-uniform guard
  int m0 = tmb * 64, n0 = tn * 16;
  int r = lane & 15, g = lane >> 4;
  const float* ap = X + (size_t)(m0 + r) * K + 2 * g;
  const float* bp = W + (size_t)(n0 + r) * K + 2 * g;
  v8f acc0 = {}, acc1 = {}, acc2 = {}, acc3 = {};

  v2f bc  = *(const v2f*)(bp);
  v2f a0c = *(const v2f*)(ap);
  v2f a1c = *(const v2f*)(ap + (size_t)16 * K);
  v2f a2c = *(const v2f*)(ap + (size_t)32 * K);
  v2f a3c = *(const v2f*)(ap + (size_t)48 * K);
  for (int k = 0; k < K - 4; k += 4) {
    // issue next k-step's loads before consuming current fragments
    v2f bn  = *(const v2f*)(bp + k + 4);
    v2f a0n = *(const v2f*)(ap + k + 4);
    v2f a1n = *(const v2f*)(ap + (size_t)16 * K + k + 4);
    v2f a2n = *(const v2f*)(ap + (size_t)32 * K + k + 4);
    v2f a3n = *(const v2f*)(ap + (size_t)48 * K + k + 4);
    acc0 = wmma_f32(a0c, bc, acc0);
    acc1 = wmma_f32(a1c, bc, acc1);
    acc2 = wmma_f32(a2c, bc, acc2);
    acc3 = wmma_f32(a3c, bc, acc3);
    bc = bn; a0c = a0n; a1c = a1n; a2c = a2n; a3c = a3n;
  }
  acc0 = wmma_f32(a0c, bc, acc0);
  acc1 = wmma_f32(a1c, bc, acc1);
  acc2 = wmma_f32(a2c, bc, acc2);
  acc3 = wmma_f32(a3c, bc, acc3);

  float bv = bias[n0 + r];
  store_tile(C, N, m0 +  0, n0, r, g, acc0, bv);
  store_tile(C, N, m0 + 16, n0, r, g, acc1, bv);
  store_tile(C, N, m0 + 32, n0, r, g, acc2, bv);
  store_tile(C, N, m0 + 48, n0, r, g, acc3, bv);
}

// ---- scores[b,i,j] = sum_h tanh(p_i[h]+p_j[h])*w[h], fused softmax -------
__global__ __launch_bounds__(256) void attn_scores(const float* __restrict__ P,
                                                   const float* __restrict__ Wa2,
                                                   const float* __restrict__ ba2,
                                                   const unsigned char* __restrict__ mask,
                                                   float* __restrict__ Aout) {
  __shared__ float pi[H];
  __shared__ float wv[H];
  __shared__ float red[8];
  int b = blockIdx.x / S;
  int i = blockIdx.x % S;
  int t = threadIdx.x;
  const float* Pb = P + (size_t)b * S * H;
  for (int h = t; h < H; h += 256) { pi[h] = Pb[(size_t)i * H + h]; wv[h] = Wa2[h]; }
  __syncthreads();

  int j = t;
  const float* pj = Pb + (size_t)j * H;
  const float4* pi4 = (const float4*)pi;     // LDS reads as ds_load_b128
  const float4* wv4 = (const float4*)wv;
  float acc = 0.f;
  for (int h4 = 0; h4 < H / 4; ++h4) {
    float4 pv = *(const float4*)(pj + 4 * h4);
    float4 pr = pi4[h4];
    float4 wr = wv4[h4];
    acc += fast_tanh(pr.x + pv.x) * wr.x;
    acc += fast_tanh(pr.y + pv.y) * wr.y;
    acc += fast_tanh(pr.z + pv.z) * wr.z;
    acc += fast_tanh(pr.w + pv.w) * wr.w;
  }
  float score = acc + ba2[0];

  int lane = t & 31, wave = t >> 5;
  // row max over ALL j (reference applies only the row mask to scores)
  float wm = warpMax(score);
  if (lane == 0) red[wave] = wm;
  __syncthreads();
  float mx = red[0];
  #pragma unroll
  for (int w = 1; w < 8; ++w) mx = fmaxf(mx, red[w]);
  float e = __expf(score - mx);
  float ez = mask[b * S + j] ? 0.f : e;   // Z excludes masked columns only
  __syncthreads();
  float ws_ = warpSum(ez);
  if (lane == 0) red[wave] = ws_;
  __syncthreads();
  float z = 0.f;
  #pragma unroll
  for (int w = 0; w < 8; ++w) z += red[w];

  bool rowMasked = mask[b * S + i] != 0;
  Aout[((size_t)b * S + i) * S + j] = rowMasked ? 0.f : e / z;
}

// ---- hid0[b] = A[b] @ hid[b]; 16x64 C-strip per wave, pipelined ----------
__global__ __launch_bounds__(256) void gemm_Ahid(const float* __restrict__ Aatt,
                                                 const float* __restrict__ Hid,
                                                 float* __restrict__ Hid0) {
  int wave = threadIdx.x >> 5, lane = threadIdx.x & 31;
  const int tilesM = S / 16, nblocks = H / 64;        // 16, 8
  int tile = blockIdx.x * 8 + wave;                   // 8*16*8 = 1024 tiles
  int b = tile / (tilesM * nblocks);
  int t2 = tile % (tilesM * nblocks);
  int tm = t2 / nblocks, tnb = t2 % nblocks;
  int m0 = tm * 16, n0 = tnb * 64;
  int r = lane & 15, g = lane >> 4;
  const float* Ab = Aatt + (size_t)b * S * S;
  const float* Hb = Hid + (size_t)b * S * H;
  const float* ap = Ab + (size_t)(m0 + r) * S + 2 * g;
  v8f acc0 = {}, acc1 = {}, acc2 = {}, acc3 = {};

  v2f ac, b0c, b1c, b2c, b3c;
  {
    ac = *(const v2f*)(ap);
    const float* h0 = Hb + (size_t)(2 * g) * H + r;
    const float* h1 = Hb + (size_t)(2 * g + 1) * H + r;
    b0c.x = h0[n0 +  0]; b0c.y = h1[n0 +  0];
    b1c.x = h0[n0 + 16]; b1c.y = h1[n0 + 16];
    b2c.x = h0[n0 + 32]; b2c.y = h1[n0 + 32];
    b3c.x = h0[n0 + 48]; b3c.y = h1[n0 + 48];
  }
  for (int k = 0; k < S - 4; k += 4) {
    v2f an = *(const v2f*)(ap + k + 4);
    const float* h0 = Hb + (size_t)(k + 4 + 2 * g) * H + r;
    const float* h1 = Hb + (size_t)(k + 4 + 2 * g + 1) * H + r;
    v2f b0n, b1n, b2n, b3n;
    b0n.x = h0[n0 +  0]; b0n.y = h1[n0 +  0];
    b1n.x = h0[n0 + 16]; b1n.y = h1[n0 + 16];
    b2n.x = h0[n0 + 32]; b2n.y = h1[n0 + 32];
    b3n.x = h0[n0 + 48]; b3n.y = h1[n0 + 48];
    acc0 = wmma_f32(ac, b0c, acc0);
    acc1 = wmma_f32(ac, b1c, acc1);
    acc2 = wmma_f32(ac, b2c, acc2);
    acc3 = wmma_f32(ac, b3c, acc3);
    ac = an; b0c = b0n; b1c = b1n; b2c = b2n; b3c = b3n;
  }
  acc0 = wmma_f32(ac, b0c, acc0);
  acc1 = wmma_f32(ac, b1c, acc1);
  acc2 = wmma_f32(ac, b2c, acc2);
  acc3 = wmma_f32(ac, b3c, acc3);

  float* Cb = Hid0 + (size_t)b * S * H;
  store_tile(Cb, H, m0, n0 +  0, r, g, acc0, 0.f);
  store_tile(Cb, H, m0, n0 + 16, r, g, acc1, 0.f);
  store_tile(Cb, H, m0, n0 + 32, r, g, acc2, 0.f);
  store_tile(Cb, H, m0, n0 + 48, r, g, acc3, 0.f);
}

// ---- dual-GEMM + gated update: hid = g*tanh(h) + (1-g)*hid ---------------
// 32x16 strip per wave, both weight fragments shared; 1-deep pipeline.
__global__ __launch_bounds__(256) void fused_update(const float* __restrict__ Hid0,
                                                    const float* __restrict__ Wm,
                                                    const float* __restrict__ bm,
                                                    const float* __restrict__ Wg,
                                                    const float* __restrict__ bg,
                                                    const unsigned char* __restrict__ mask,
                                                    float* __restrict__ Hid) {
  int wave = threadIdx.x >> 5, lane = threadIdx.x & 31;
  const int tilesN = H / 16;                 // M = B*S = 2048 rows
  int tile = blockIdx.x * 8 + wave;          // over (M/32)*tilesN = 2048
  int tmb = tile / tilesN, tn = tile % tilesN;
  int m0 = tmb * 32, n0 = tn * 16;
  int r = lane & 15, g = lane >> 4;
  const float* ap  = Hid0 + (size_t)(m0 + r) * H + 2 * g;
  const float* bmp = Wm + (size_t)(n0 + r) * H + 2 * g;
  const float* bgp = Wg + (size_t)(n0 + r) * H + 2 * g;
  v8f accm0 = {}, accg0 = {}, accm1 = {}, accg1 = {};

  v2f a0c = *(const v2f*)(ap);
  v2f a1c = *(const v2f*)(ap + (size_t)16 * H);
  v2f w1c = *(const v2f*)(bmp);
  v2f w2c = *(const v2f*)(bgp);
  for (int k = 0; k < H - 4; k += 4) {
    v2f a0n = *(const v2f*)(ap + k + 4);
    v2f a1n = *(const v2f*)(ap + (size_t)16 * H + k + 4);
    v2f w1n = *(const v2f*)(bmp + k + 4);
    v2f w2n = *(const v2f*)(bgp + k + 4);
    accm0 = wmma_f32(a0c, w1c, accm0);
    accg0 = wmma_f32(a0c, w2c, accg0);
    accm1 = wmma_f32(a1c, w1c, accm1);
    accg1 = wmma_f32(a1c, w2c, accg1);
    a0c = a0n; a1c = a1n; w1c = w1n; w2c = w2n;
  }
  accm0 = wmma_f32(a0c, w1c, accm0);
  accg0 = wmma_f32(a0c, w2c, accg0);
  accm1 = wmma_f32(a1c, w1c, accm1);
  accg1 = wmma_f32(a1c, w2c, accg1);

  float bmv = bm[n0 + r], bgv = bg[n0 + r];
  #pragma unroll
  for (int sub = 0; sub < 2; ++sub) {
    v8f am = sub ? accm1 : accm0;
    v8f ag = sub ? accg1 : accg0;
    #pragma unroll
    for (int v = 0; v < 8; ++v) {
      int m = m0 + sub * 16 + v + 8 * g;     // m = b*S + i : flat mask index
      float hval = fast_tanh(am[v] + bmv);
      float gval = 1.f / (1.f + __expf(-(ag[v] + bgv)));
      if (mask[m]) hval = 0.f;
      size_t idx = (size_t)m * H + n0 + r;
      float old = Hid[idx];
      Hid[idx] = gval * hval + (1.f - gval) * old;
    }
  }
}

extern "C" void kernel_launch(void* const* d_in, const int* in_sizes, int n_in,
                              void* d_out, int out_size, void* d_ws, size_t ws_size,
                              hipStream_t stream) {
  const float* input = (const float*)d_in[0];
  const unsigned char* mask = (const unsigned char*)d_in[1]; // jnp bool = 1 byte
  const float* W_a1 = (const float*)d_in[2];
  const float* b_a1 = (const float*)d_in[3];
  const float* W_a2 = (const float*)d_in[4];
  const float* b_a2 = (const float*)d_in[5];
  const float* W_m  = (const float*)d_in[6];
  const float* b_m  = (const float*)d_in[7];
  const float* W_g  = (const float*)d_in[8];
  const float* b_g  = (const float*)d_in[9];
  float* out = (float*)d_out;

  float* P    = (float*)d_ws;                 // [B,S,H]  4 MB
  float* Aatt = P + (size_t)B * S * H;        // [B,S,S]  2 MB
  float* hid  = Aatt + (size_t)B * S * S;     // [B,S,H]  4 MB
  float* hid0 = out;                          // [B,S,H] aliases d_out (fully
                                              // rewritten every iteration)

  const int ELT_BLOCKS  = (B * S * H / 4) / 256;            // 1024
  const int XWT_BLOCKS  = ((B * S / 64) * (H / 16)) / 8;    // 128
  const int AHID_BLOCKS = (B * (S / 16) * (H / 64)) / 8;    // 128
  const int UPD_BLOCKS  = ((B * S / 32) * (H / 16)) / 8;    // 256

  transpose_in<<<ELT_BLOCKS, 256, 0, stream>>>(input, hid);
  gemm_xwT<<<XWT_BLOCKS, 256, 0, stream>>>(hid, W_a1, b_a1, P, B * S, H, H);
  attn_scores<<<B * S, 256, 0, stream>>>(P, W_a2, b_a2, mask, Aatt);
  for (int it = 0; it < NITER; ++it) {
    gemm_Ahid<<<AHID_BLOCKS, 256, 0, stream>>>(Aatt, hid, hid0);
    fused_update<<<UPD_BLOCKS, 256, 0, stream>>>(hid0, W_m, b_m, W_g, b_g, mask, hid);
  }
  transpose_out<<<ELT_BLOCKS, 256, 0, stream>>>(hid, out);
}